// SoftmaxGHMc_38671885533681
// MI455X (gfx1250) — compile-verified
//
#include <hip/hip_runtime.h>
#include <stdint.h>

// ---------------------------------------------------------------------------
// GHM-C loss for pred[N=131072, C=1000] f32, target[N] i64 -> scalar f32.
// Memory-bound: 524 MB of pred streamed once => ~22.5us floor @ 23.3 TB/s.
// CDNA5 paths: async global->LDS DMA (ASYNCcnt) with per-wave double
// buffering + non-temporal hints (524MB stream >> 192MB L2, zero reuse).
// No GEMM structure (AI ~1.5 flop/byte) => WMMA not applicable by roofline.
// ---------------------------------------------------------------------------

#define WAVES_PER_WG 8
#define ROW_PAD_FLOATS 1024      // C=1000 padded to 1024 floats (4096 B)
#define ROW_PAD_BYTES 4096
#define WSUM_BLOCKS 256

// 8 async 512B wave-issues copy one (padded) row into LDS.
// GVS mode: global = SADDR(row) + VADDR(lane*16) + imm ; LDS = VDST + imm.
// (CDNA5 ISA: INST_OFFSET is added to BOTH addresses.)
// th:TH_LOAD_NT — single-use stream, don't pollute L2.
__device__ __forceinline__ void async_row_copy(unsigned lds_addr, unsigned voff,
                                               unsigned voff_tail,
                                               const float* row) {
  asm volatile(
      "global_load_async_to_lds_b128 %0, %1, %3 offset:0 th:TH_LOAD_NT\n\t"
      "global_load_async_to_lds_b128 %0, %1, %3 offset:512 th:TH_LOAD_NT\n\t"
      "global_load_async_to_lds_b128 %0, %1, %3 offset:1024 th:TH_LOAD_NT\n\t"
      "global_load_async_to_lds_b128 %0, %1, %3 offset:1536 th:TH_LOAD_NT\n\t"
      "global_load_async_to_lds_b128 %0, %1, %3 offset:2048 th:TH_LOAD_NT\n\t"
      "global_load_async_to_lds_b128 %0, %1, %3 offset:2560 th:TH_LOAD_NT\n\t"
      "global_load_async_to_lds_b128 %0, %1, %3 offset:3072 th:TH_LOAD_NT\n\t"
      "global_load_async_to_lds_b128 %0, %2, %3 offset:3584 th:TH_LOAD_NT\n\t"
      :
      : "v"(lds_addr), "v"(voff), "v"(voff_tail), "s"(row)
      : "memory");
}

__device__ __forceinline__ void wait_async_le8() {
  asm volatile("s_wait_asynccnt 0x8" ::: "memory");
}
__device__ __forceinline__ void wait_async_0() {
  asm volatile("s_wait_asynccnt 0x0" ::: "memory");
}

// bin = searchsorted(edges, vc, 'right') - 1, edges = arange(11)/10 in f32.
// vc in [0,1]; literal edges round identically to float(j)/10.
__device__ __forceinline__ int ghm_bin(float vc) {
  int b = 0;
  b += (vc >= 0.1f); b += (vc >= 0.2f); b += (vc >= 0.3f);
  b += (vc >= 0.4f); b += (vc >= 0.5f); b += (vc >= 0.6f);
  b += (vc >= 0.7f); b += (vc >= 0.8f); b += (vc >= 0.9f);
  return b;  // 0..9 ; vc==1.0 lands in bin 9 like the nudged last edge
}

// ---------------------------------------------------------------------------
__global__ void ghm_init_kernel(unsigned* maxbits, unsigned* counts) {
  if (threadIdx.x == 0) *maxbits = 0u;
  if (threadIdx.x < 10) counts[threadIdx.x] = 0u;
}

// ---------------------------------------------------------------------------
// One wave per row; async double-buffered row streaming through LDS.
__global__ __launch_bounds__(WAVES_PER_WG * 32) void ghm_rowce_kernel(
    const float* __restrict__ pred, const long long* __restrict__ tgt,
    float* __restrict__ values, unsigned* __restrict__ maxbits, int N, int C) {
  __shared__ float smem[WAVES_PER_WG * 2 * ROW_PAD_FLOATS];  // 64 KB

  const int lane = threadIdx.x & 31;
  const int wave = threadIdx.x >> 5;
  const int gwave = blockIdx.x * WAVES_PER_WG + wave;
  const int nwave = gridDim.x * WAVES_PER_WG;

  float* wbuf = &smem[wave * 2 * ROW_PAD_FLOATS];
  // generic-pointer low 32 bits == LDS byte offset on AMDGCN
  unsigned lds0 = (unsigned)(uint64_t)(void*)wbuf + (unsigned)lane * 16u;
  unsigned lds1 = lds0 + ROW_PAD_BYTES;
  unsigned voff = (unsigned)lane * 16u;
  // last 512B chunk: lanes past the row end are clamped in-bounds (garbage
  // lands in padding elements [C..1023], masked to -inf at consume time;
  // elements [0..C) are always written correctly)
  int tail_lanes = (C * 4 - 3584) >> 4;  // =26 for C=1000
  unsigned voff_tail = (lane < tail_lanes) ? voff : 0u;

  int r = gwave;
  if (r < N) async_row_copy(lds0, voff, voff_tail, pred + (size_t)r * C);
  int buf = 0;
  for (; r < N; r += nwave) {
    int rn = r + nwave;
    if (rn < N) {
      async_row_copy(buf ? lds0 : lds1, voff, voff_tail,
                     pred + (size_t)rn * C);
      wait_async_le8();  // 16 outstanding -> oldest 8 (current row) are done
    } else {
      wait_async_0();
    }

    const float* rowlds = &smem[(wave * 2 + buf) * ROW_PAD_FLOATS];
    const float4* p = (const float4*)rowlds;
    float4 v[8];
#pragma unroll
    for (int j = 0; j < 8; ++j) v[j] = p[j * 32 + lane];
#pragma unroll
    for (int j = 0; j < 8; ++j) {
      int base = (j * 32 + lane) * 4;
      if (base + 0 >= C) v[j].x = -__builtin_inff();
      if (base + 1 >= C) v[j].y = -__builtin_inff();
      if (base + 2 >= C) v[j].z = -__builtin_inff();
      if (base + 3 >= C) v[j].w = -__builtin_inff();
    }

    // pass 1: row max
    float m = -__builtin_inff();
#pragma unroll
    for (int j = 0; j < 8; ++j)
      m = fmaxf(m, fmaxf(fmaxf(v[j].x, v[j].y), fmaxf(v[j].z, v[j].w)));
#pragma unroll
    for (int off = 16; off > 0; off >>= 1)
      m = fmaxf(m, __shfl_xor(m, off, 32));

    // pass 2: sum exp(x - m)  (masked entries: exp(-inf)=0)
    float s = 0.f;
#pragma unroll
    for (int j = 0; j < 8; ++j)
      s += __expf(v[j].x - m) + __expf(v[j].y - m) + __expf(v[j].z - m) +
           __expf(v[j].w - m);
#pragma unroll
    for (int off = 16; off > 0; off >>= 1) s += __shfl_xor(s, off, 32);

    if (lane == 0) {
      long long t = tgt[r];
      float pt = rowlds[(int)t];  // row already in LDS: ds_load, no global trip
      float val = m + __logf(s) - pt;  // -log_softmax[target] >= 0
      values[r] = val;
      // float bits are order-preserving for non-negative floats
      atomicMax(maxbits, __float_as_uint(val));
    }
    buf ^= 1;
  }
}

// ---------------------------------------------------------------------------
__global__ void ghm_hist_kernel(const float* __restrict__ values,
                                const unsigned* __restrict__ maxbits,
                                unsigned* __restrict__ counts, int N) {
  __shared__ unsigned h[10];
  if (threadIdx.x < 10) h[threadIdx.x] = 0u;
  __syncthreads();
  float vmax = __uint_as_float(*maxbits);
  int i = blockIdx.x * blockDim.x + threadIdx.x;
  int stride = gridDim.x * blockDim.x;
  for (; i < N; i += stride) {
    float vc = values[i] / vmax;
    atomicAdd(&h[ghm_bin(vc)], 1u);  // integer atomics: order-independent
  }
  __syncthreads();
  if (threadIdx.x < 10) atomicAdd(&counts[threadIdx.x], h[threadIdx.x]);
}

// ---------------------------------------------------------------------------
// Stage 1 of deterministic weighted sum: each block writes its own partial
// (no float atomics -> bitwise reproducible across graph replays).
__global__ __launch_bounds__(256) void ghm_wsum_kernel(
    const float* __restrict__ values, const unsigned* __restrict__ maxbits,
    const unsigned* __restrict__ counts, float* __restrict__ partial, int N) {
  __shared__ float red[256];
  float vmax = __uint_as_float(*maxbits);
  float tot = (float)N;
  float nn = 0.f;
#pragma unroll
  for (int j = 0; j < 10; ++j) nn += (counts[j] > 0u) ? 1.f : 0.f;
  nn = fmaxf(nn, 1.f);
  float wbin[10];
#pragma unroll
  for (int j = 0; j < 10; ++j) {
    unsigned cj = counts[j];
    float c = (cj > 0u) ? (float)cj : 1.f;
    wbin[j] = ((tot / c) / nn) / tot;  // match reference op order
  }
  float acc = 0.f;
  for (int i = blockIdx.x * blockDim.x + threadIdx.x; i < N;
       i += gridDim.x * blockDim.x) {
    float v = values[i];
    acc += wbin[ghm_bin(v / vmax)] * v;
  }
  red[threadIdx.x] = acc;
  __syncthreads();
  for (int s = 128; s > 0; s >>= 1) {
    if ((int)threadIdx.x < s) red[threadIdx.x] += red[threadIdx.x + s];
    __syncthreads();
  }
  if (threadIdx.x == 0) partial[blockIdx.x] = red[0];
}

// Stage 2: fold the fixed number of partials in a fixed order.
__global__ __launch_bounds__(256) void ghm_out_kernel(
    const float* __restrict__ partial, float* __restrict__ out, int nb) {
  __shared__ float red[256];
  float a = ((int)threadIdx.x < nb) ? partial[threadIdx.x] : 0.f;
  red[threadIdx.x] = a;
  __syncthreads();
  for (int s = 128; s > 0; s >>= 1) {
    if ((int)threadIdx.x < s) red[threadIdx.x] += red[threadIdx.x + s];
    __syncthreads();
  }
  if (threadIdx.x == 0) out[0] = red[0];
}

// ---------------------------------------------------------------------------
extern "C" void kernel_launch(void* const* d_in, const int* in_sizes, int n_in,
                              void* d_out, int out_size, void* d_ws,
                              size_t ws_size, hipStream_t stream) {
  const float* pred = (const float*)d_in[0];
  const long long* tgt = (const long long*)d_in[1];  // reference dtype: int64
  float* out = (float*)d_out;

  int NC = in_sizes[0];
  int N = in_sizes[1];
  int C = NC / N;  // 1000

  unsigned char* ws = (unsigned char*)d_ws;
  unsigned* maxbits = (unsigned*)ws;                  // [0]  float bits of max
  unsigned* counts = (unsigned*)(ws + 4);             // [10] bin counts
  float* values = (float*)(ws + 64);                  // [N]  per-sample CE
  float* partial = (float*)(ws + 64 + (size_t)N * 4); // [WSUM_BLOCKS]

  ghm_init_kernel<<<1, 64, 0, stream>>>(maxbits, counts);
  ghm_rowce_kernel<<<1024, WAVES_PER_WG * 32, 0, stream>>>(pred, tgt, values,
                                                           maxbits, N, C);
  ghm_hist_kernel<<<256, 256, 0, stream>>>(values, maxbits, counts, N);
  ghm_wsum_kernel<<<WSUM_BLOCKS, 256, 0, stream>>>(values, maxbits, counts,
                                                   partial, N);
  ghm_out_kernel<<<1, 256, 0, stream>>>(partial, out, WSUM_BLOCKS);
}